// ReconstructionLoss_34385508171850
// MI455X (gfx1250) — compile-verified
//
#include <hip/hip_runtime.h>

// Problem constants (from the reference).
#define MS      4
#define BATCH   4
#define SEQ     512
#define VOCAB_N 32000
#define SM1     511                 // SEQ - 1 positions that contribute
#define ROWS    (MS * BATCH * SM1)  // 8176 logsumexp rows
#define TPB     256                 // 8 wave32 per workgroup
#define V4      (VOCAB_N / 4)       // 8000 float4 per row
#define SLICE_ELEMS (VOCAB_N / 8)   // 4000 fp32 per wave's TDM slice
#define SLICE_BYTES (SLICE_ELEMS * 4) // 16000 B

typedef unsigned __attribute__((ext_vector_type(4))) uint4v;
typedef int      __attribute__((ext_vector_type(8))) int8v;
typedef int      __attribute__((ext_vector_type(4))) int4v;

// ---------------------------------------------------------------------------
// Kernel 1: per-row logsumexp + target gather.
// One workgroup per (step, batch, seq) row. The 128 KB fp32 row is staged
// HBM -> LDS exactly once by the Tensor Data Mover (one descriptor per wave,
// 16 KB contiguous slice each), then both softmax passes and the target
// gather run out of LDS.
// ---------------------------------------------------------------------------
__global__ __launch_bounds__(TPB) void row_gather_lse(
    const float* __restrict__ y_pred,   // [MS, BATCH, SEQ, VOCAB_N]
    const int*   __restrict__ y_true,   // [BATCH, SEQ]
    const int*   __restrict__ pad_id_p, // scalar on device
    float*       __restrict__ row_out)  // [ROWS]
{
    extern __shared__ __align__(16) float smem[];  // VOCAB_N floats = 128000 B
    __shared__ float red[TPB];

    const int r    = blockIdx.x;
    const int tid  = threadIdx.x;
    const int step = r / (BATCH * SM1);
    const int rem  = r % (BATCH * SM1);
    const int b    = rem / SM1;
    const int s    = rem % SM1;        // 0 .. 510

    const float* row =
        y_pred + (((size_t)step * BATCH + (size_t)b) * SEQ + (size_t)s) * (size_t)VOCAB_N;

    // ---- TDM stage: HBM -> LDS, one 16 KB descriptor per wave -------------
    const unsigned wid = __builtin_amdgcn_readfirstlane((unsigned)tid >> 5);
    const unsigned lds_off =
        (unsigned)(size_t)(void*)smem + wid * (unsigned)SLICE_BYTES;
    const unsigned long long ga =
        (unsigned long long)(size_t)(const void*)row +
        (unsigned long long)wid * (unsigned long long)SLICE_BYTES;

    // D# group 0: count=1 | lds_addr | global_addr[56:0] | type=2
    uint4v g0;
    g0[0] = 1u;                                    // count=1 (valid user D#)
    g0[1] = lds_off;                               // lds_addr (bytes)
    g0[2] = (unsigned)ga;                          // global_addr[31:0]
    g0[3] = (unsigned)((ga >> 32) & 0x01FFFFFFull) // global_addr[56:32]
            | (2u << 30);                          // type=2 ("image")

    // D# group 1: data_size=4B; 1-D tile of SLICE_ELEMS elements
    int8v g1;
    g1[0] = (int)(2u << 16);                           // data_size=2 (4B); mask/pad/iter=0
    g1[1] = (int)(((unsigned)SLICE_ELEMS & 0xFFFFu) << 16);    // tensor_dim0[15:0] @63:48
    g1[2] = (int)(((unsigned)SLICE_ELEMS >> 16) | (1u << 16)); // tensor_dim0[31:16] | tensor_dim1=1
    g1[3] = (int)(((unsigned)SLICE_ELEMS & 0xFFFFu) << 16);    // tensor_dim1[31:16]=0 | tile_dim0
    g1[4] = 1;                                          // tile_dim1=1 | tile_dim2=0 (unused)
    g1[5] = (int)SLICE_ELEMS;                           // tensor_dim0_stride[31:0]
    g1[6] = 0;                                          // stride0[47:32] | dim1_stride[15:0]
    g1[7] = 0;                                          // dim1_stride[47:16]

    // D# groups 2/3: higher dims extent 1 so OOB checks cannot clip, tiles unused
    int4v g2;
    g2[0] = 1;            // tensor_dim2 = 1
    g2[1] = 1;            // tensor_dim3 = 1
    g2[2] = 0;            // tensor_dim2_stride[31:0]
    g2[3] = 0;            // stride2[47:32] | tile_dim3=0 (unused)
    int4v g3;
    g3[0] = 0;            // tensor_dim3_stride[31:0]
    g3[1] = (int)(1u << 16); // stride3[47:32]=0 | tensor_dim4[15:0]=1
    g3[2] = 0;            // tensor_dim4[31:16]=0 | tile_dim4=0 (unused)
    g3[3] = 0;            // reserved

    // Trailing v8 group: VADDR4 is architecturally unused -> zero/neutral.
    int8v g4 = {0, 0, 0, 0, 0, 0, 0, 0};

    __builtin_amdgcn_tensor_load_to_lds(g0, g1, g2, g3, g4, 0);
    __builtin_amdgcn_s_wait_tensorcnt((short)0);   // this wave's slice is in LDS
    __syncthreads();                               // all 8 slices visible

    const float4* s4 = (const float4*)smem;

    // ---- Pass 1: row max --------------------------------------------------
    float m = -__builtin_inff();
    for (int j = tid; j < V4; j += TPB) {
        float4 v = s4[j];
        m = fmaxf(m, fmaxf(fmaxf(v.x, v.y), fmaxf(v.z, v.w)));
    }
    red[tid] = m;
    for (int off = TPB / 2; off > 0; off >>= 1) {
        __syncthreads();
        if (tid < off) red[tid] = fmaxf(red[tid], red[tid + off]);
    }
    __syncthreads();
    m = red[0];
    __syncthreads();

    // ---- Pass 2: sum of exp(x - m) ---------------------------------------
    float sum = 0.0f;
    for (int j = tid; j < V4; j += TPB) {
        float4 v = s4[j];
        sum += __expf(v.x - m) + __expf(v.y - m) +
               __expf(v.z - m) + __expf(v.w - m);
    }
    red[tid] = sum;
    for (int off = TPB / 2; off > 0; off >>= 1) {
        __syncthreads();
        if (tid < off) red[tid] += red[tid + off];
    }
    __syncthreads();

    if (tid == 0) {
        float lse = m + __logf(red[0]);
        int   tgt = y_true[b * SEQ + (s + 1)];
        float val = (tgt != pad_id_p[0]) ? (smem[tgt] - lse) : 0.0f;
        row_out[r] = val;
    }
}

// ---------------------------------------------------------------------------
// Kernel 2: deterministic weighted reduction of the 8176 row values.
// loss = -(1/BATCH) * sum_{step,b,s} p[step,b] * rowval[step,b,s]
// ---------------------------------------------------------------------------
__global__ __launch_bounds__(TPB) void finalize_loss(
    const float* __restrict__ rowvals,  // [ROWS]
    const float* __restrict__ p,        // [MS, BATCH]
    float*       __restrict__ out)      // [1]
{
    __shared__ float red[TPB];
    const int tid = threadIdx.x;

    float acc = 0.0f;
    for (int r = tid; r < ROWS; r += TPB) {
        const int step = r / (BATCH * SM1);
        const int rem  = r % (BATCH * SM1);
        const int b    = rem / SM1;
        acc += p[step * BATCH + b] * rowvals[r];
    }
    red[tid] = acc;
    for (int off = TPB / 2; off > 0; off >>= 1) {
        __syncthreads();
        if (tid < off) red[tid] += red[tid + off];
    }
    __syncthreads();
    if (tid == 0) out[0] = -red[0] / (float)BATCH;
}

// ---------------------------------------------------------------------------
extern "C" void kernel_launch(void* const* d_in, const int* in_sizes, int n_in,
                              void* d_out, int out_size, void* d_ws, size_t ws_size,
                              hipStream_t stream)
{
    (void)in_sizes; (void)n_in; (void)out_size; (void)ws_size;

    const float* p      = (const float*)d_in[0];  // [MS, BATCH] fp32
    const float* y_pred = (const float*)d_in[1];  // [MS, BATCH, SEQ, VOCAB_N] fp32
    const int*   y_true = (const int*)d_in[2];    // [BATCH, SEQ] int32
    const int*   pad_id = (const int*)d_in[3];    // scalar int

    float* rowvals = (float*)d_ws;   // ROWS floats = 32704 B of scratch
    float* out     = (float*)d_out;  // 1 float

    const size_t shmem = (size_t)VOCAB_N * sizeof(float);  // 128000 B
    row_gather_lse<<<ROWS, TPB, shmem, stream>>>(y_pred, y_true, pad_id, rowvals);
    finalize_loss<<<1, TPB, 0, stream>>>(rowvals, p, out);
}